// GCN_7541962572403
// MI455X (gfx1250) — compile-verified
//
#include <hip/hip_runtime.h>

typedef __attribute__((ext_vector_type(2))) float v2f;
typedef __attribute__((ext_vector_type(8))) float v8f;

#define BSTRIDE 80  // float2 per k-pair row: +80 f2 = 160 dwords = +32 banks -> conflict-free halves

// ---------------------------------------------------------------------------
// T[N,64] = Hin[N,K] @ W[K,64] via V_WMMA_F32_16X16X4_F32 (full fp32).
// REQUIRES K % 4 == 0 (inputs are pre-padded). 256-thread block = 8 waves;
// each wave owns a 16-row block, computes all four 16-col tiles.
// A frag (ISA 7.12.2): lanes 0-15 -> K{k0,k0+1}, lanes 16-31 -> K{k0+2,k0+3},
// VGPR0 = first of pair -> one global_load_b64 per step.
// B frag mirrors the half-wave K split; W staged in LDS as interleaved float2
// (W[2kp][n], W[2kp+1][n]) so each fragment is one ds_load_b64.
// ---------------------------------------------------------------------------
__global__ __launch_bounds__(256) void gemm_wmma_f32(
    const float* __restrict__ Hin, int K,     // K % 4 == 0, K <= 128
    const float* __restrict__ W,              // [K,64] row-major
    float* __restrict__ T,                    // [N,64]
    int N)
{
  __shared__ float2 sB[64 * BSTRIDE];         // 64 k-pairs max (K=128): 40 KB
  const int tid = threadIdx.x;
  const int Kp2 = K >> 1;
  for (int idx = tid; idx < (Kp2 << 6); idx += 256) {
    int kp = idx >> 6, n = idx & 63;
    sB[kp * BSTRIDE + n] = make_float2(W[(2 * kp) * 64 + n], W[(2 * kp + 1) * 64 + n]);
  }
  __syncthreads();

  const int lane = tid & 31;
  const int m    = lane & 15;
  const int hi   = lane >> 4;                 // 0: K pair {0,1}; 1: K pair {2,3}
  const int row0 = blockIdx.x * 128 + (tid >> 5) * 16;
  if (row0 >= N) return;                      // N multiple of 16 per-wave

  const float*  arow = Hin + (size_t)(row0 + m) * (size_t)K + 2 * hi;
  const float2* brow = sB + hi * BSTRIDE + m;

  v8f acc0 = {}, acc1 = {}, acc2 = {}, acc3 = {};
  for (int k0 = 0; k0 < K; k0 += 4) {
    const v2f a = *(const v2f*)(arow + k0);            // global_load_b64
    const float2* b = brow + (k0 >> 1) * BSTRIDE;
    const v2f b0 = *(const v2f*)(b +  0);              // ds_load_b64 x4
    const v2f b1 = *(const v2f*)(b + 16);
    const v2f b2 = *(const v2f*)(b + 32);
    const v2f b3 = *(const v2f*)(b + 48);
    acc0 = __builtin_amdgcn_wmma_f32_16x16x4_f32(false, a, false, b0, (short)0, acc0, false, false);
    acc1 = __builtin_amdgcn_wmma_f32_16x16x4_f32(false, a, false, b1, (short)0, acc1, false, false);
    acc2 = __builtin_amdgcn_wmma_f32_16x16x4_f32(false, a, false, b2, (short)0, acc2, false, false);
    acc3 = __builtin_amdgcn_wmma_f32_16x16x4_f32(false, a, false, b3, (short)0, acc3, false, false);
  }

  // C/D layout: VGPR r -> row (hi*8 + r), col = tile*16 + m
  float* trow = T + (size_t)row0 * 64 + m;
#pragma unroll
  for (int r = 0; r < 8; ++r) {
    const int rr = r + hi * 8;
    trow[rr * 64 +  0] = acc0[r];
    trow[rr * 64 + 16] = acc1[r];
    trow[rr * 64 + 32] = acc2[r];
    trow[rr * 64 + 48] = acc3[r];
  }
}

// Pad x[N,IN] -> xp[N,KP] with zeros so the GEMM never needs K guards.
__global__ void pad_input(const float* __restrict__ x, float* __restrict__ xp,
                          int N, int IN, int KP) {
  int i = blockIdx.x * 256 + threadIdx.x;
  if (i >= N * KP) return;
  int node = i / KP, j = i - node * KP;
  xp[i] = (j < IN) ? x[node * IN + j] : 0.0f;
}

// --------------------------- degree / norm ---------------------------------
__global__ void fill1(float* __restrict__ p, int n) {
  int i = blockIdx.x * 256 + threadIdx.x;
  if (i < n) p[i] = 1.0f;                     // self-loop contributes 1
}
__global__ void deg_edges(const int* __restrict__ ecol, float* __restrict__ deg, int E) {
  int e = blockIdx.x * 256 + threadIdx.x;
  if (e < E) atomicAdd(&deg[ecol[e]], 1.0f);
}
__global__ void rsqrt_inplace(float* __restrict__ p, int n) {
  int i = blockIdx.x * 256 + threadIdx.x;
  if (i < n) p[i] = rsqrtf(p[i]);             // deg >= 1 always (self-loops)
}

// --------------------------- conv scatter ----------------------------------
__global__ void selfloop_init(const float* __restrict__ T, const float* __restrict__ dinv,
                              float* __restrict__ Acc, int total) {
  int i = blockIdx.x * 256 + threadIdx.x;
  if (i >= total) return;
  float d = dinv[i >> 6];
  Acc[i] = d * d * T[i];
}

// One wave per edge: coalesced 256B gather of t[row], 64 contiguous f32 atomics.
__global__ __launch_bounds__(256) void scatter_edges(
    const int* __restrict__ erow, const int* __restrict__ ecol,
    const float* __restrict__ dinv, const float* __restrict__ T,
    float* __restrict__ Acc, int E)
{
  const int e = blockIdx.x * 8 + (threadIdx.x >> 5);
  if (e >= E) return;
  const int lane = threadIdx.x & 31;
  const int r = erow[e];
  const int c = ecol[e];
  const float nrm = dinv[r] * dinv[c];
  const float2 v = *(const float2*)(T + (size_t)r * 64 + 2 * lane);
  float* dst = Acc + (size_t)c * 64 + 2 * lane;
  atomicAdd(dst + 0, nrm * v.x);
  atomicAdd(dst + 1, nrm * v.y);
}

__global__ void bias_bn_relu(float* __restrict__ Acc, const float* __restrict__ bias,
                             const float* __restrict__ g,  const float* __restrict__ bb,
                             const float* __restrict__ mm, const float* __restrict__ vv,
                             int total) {
  int i = blockIdx.x * 256 + threadIdx.x;
  if (i >= total) return;
  int j = i & 63;
  float s = Acc[i] + bias[j];
  s = (s - mm[j]) * rsqrtf(vv[j] + 1e-5f) * g[j] + bb[j];
  Acc[i] = fmaxf(s, 0.0f);
}

// ----------------------------- pooling -------------------------------------
__global__ void pool_zero(float* __restrict__ gsum, unsigned* __restrict__ gmax,
                          float* __restrict__ gcnt, int G) {
  int i = blockIdx.x * 256 + threadIdx.x;
  if (i < G * 64) { gsum[i] = 0.0f; gmax[i] = 0u; }   // h>=0 post-ReLU -> max init 0
  if (i < G) gcnt[i] = 0.0f;
}
__global__ void pool_acc(const float* __restrict__ H, const int* __restrict__ batch,
                         float* __restrict__ gsum, unsigned* __restrict__ gmax,
                         float* __restrict__ gcnt, int total) {
  int i = blockIdx.x * 256 + threadIdx.x;
  if (i >= total) return;
  int node = i >> 6, j = i & 63;
  int g = batch[node];
  float v = H[i];
  atomicAdd(&gsum[g * 64 + j], v);
  atomicMax(&gmax[g * 64 + j], __float_as_uint(v));   // bit-order == float-order for v>=0
  if (j == 0) atomicAdd(&gcnt[g], 1.0f);
}
__global__ void pool_final(const float* __restrict__ gsum, const unsigned* __restrict__ gmax,
                           const float* __restrict__ gcnt, float* __restrict__ pooled, int G) {
  int i = blockIdx.x * 256 + threadIdx.x;
  if (i >= G * 64) return;
  int g = i >> 6, j = i & 63;
  float cnt = fmaxf(gcnt[g], 1.0f);
  pooled[g * 128 + j]      = gsum[i] / cnt;
  pooled[g * 128 + 64 + j] = __uint_as_float(gmax[i]);
}

// ---------------------------- classifier -----------------------------------
__global__ void bias_relu(float* __restrict__ X, const float* __restrict__ b, int total) {
  int i = blockIdx.x * 256 + threadIdx.x;
  if (i >= total) return;
  X[i] = fmaxf(X[i] + b[i & 63], 0.0f);
}
__global__ void mlp_out(const float* __restrict__ hid, const float* __restrict__ Wc2,
                        const float* __restrict__ bc2, float* __restrict__ out, int G, int C) {
  int i = blockIdx.x * 256 + threadIdx.x;
  if (i >= G * C) return;
  int g = i / C, c = i - g * C;
  float s = bc2[c];
  const float* h = hid + (size_t)g * 64;
  for (int k = 0; k < 64; ++k) s += h[k] * Wc2[k * C + c];
  out[i] = s;
}

// ---------------------------------------------------------------------------
extern "C" void kernel_launch(void* const* d_in, const int* in_sizes, int n_in,
                              void* d_out, int out_size, void* d_ws, size_t ws_size,
                              hipStream_t stream) {
  (void)n_in; (void)ws_size;
  const float* x     = (const float*)d_in[0];
  const int*   erow  = (const int*)d_in[1];
  const int*   ecol  = (const int*)d_in[2];
  const int*   batch = (const int*)d_in[3];
  const float* W0    = (const float*)d_in[4];
  const float* b0    = (const float*)d_in[5];
  const float* W1    = (const float*)d_in[6];
  const float* b1    = (const float*)d_in[7];
  const float* W2    = (const float*)d_in[8];
  const float* b2    = (const float*)d_in[9];
  const float* bn_g  = (const float*)d_in[10];
  const float* bn_b  = (const float*)d_in[11];
  const float* bn_m  = (const float*)d_in[12];
  const float* bn_v  = (const float*)d_in[13];
  const float* Wc1   = (const float*)d_in[14];
  const float* bc1   = (const float*)d_in[15];
  const float* Wc2   = (const float*)d_in[16];
  const float* bc2   = (const float*)d_in[17];

  const int N  = in_sizes[3];
  const int E  = in_sizes[1];
  const int IN = in_sizes[0] / N;
  const int C  = in_sizes[17];
  const int G  = out_size / C;
  const int KP = (IN + 3) & ~3;            // padded input width (26 -> 28)

  // Workspace layout (~64 MB): two N x 64 feature buffers ping-pong in-place,
  // plus padded input, dinv, and small pooling/classifier buffers.
  float*    ws     = (float*)d_ws;
  size_t    NH     = (size_t)N * 64;
  float*    X      = ws;                       // feature buffer (conv output)
  float*    Y      = X + NH;                   // transformed features h@W
  float*    dinv   = Y + NH;                   // N
  float*    gsum   = dinv + N;                 // G*64
  unsigned* gmax   = (unsigned*)(gsum + (size_t)G * 64);
  float*    gcnt   = (float*)(gmax + (size_t)G * 64);
  float*    pooled = gcnt + G;                 // G*128
  float*    tmp1   = pooled + (size_t)G * 128; // G*64
  float*    xpad   = tmp1 + (size_t)G * 64;    // N*KP

  auto cdiv = [](int a, int b) { return (a + b - 1) / b; };
  const int NHi = (int)NH;

  // zero-pad layer-0 input so the GEMM K is always a multiple of 4
  pad_input<<<cdiv(N * KP, 256), 256, 0, stream>>>(x, xpad, N, IN, KP);

  // degree + D^-1/2 (self-loops included)
  fill1<<<cdiv(N, 256), 256, 0, stream>>>(dinv, N);
  deg_edges<<<cdiv(E, 256), 256, 0, stream>>>(ecol, dinv, E);
  rsqrt_inplace<<<cdiv(N, 256), 256, 0, stream>>>(dinv, N);
  pool_zero<<<cdiv(G * 64, 256), 256, 0, stream>>>(gsum, gmax, gcnt, G);

  const float* hin = xpad;
  int K = KP;
  const float* Ws[3] = {W0, W1, W2};
  const float* bs[3] = {b0, b1, b2};
  for (int l = 0; l < 3; ++l) {
    gemm_wmma_f32<<<cdiv(N, 128), 256, 0, stream>>>(hin, K, Ws[l], Y, N);
    selfloop_init<<<cdiv(NHi, 256), 256, 0, stream>>>(Y, dinv, X, NHi);
    scatter_edges<<<cdiv(E, 8), 256, 0, stream>>>(erow, ecol, dinv, Y, X, E);
    bias_bn_relu<<<cdiv(NHi, 256), 256, 0, stream>>>(
        X, bs[l], bn_g + l * 64, bn_b + l * 64, bn_m + l * 64, bn_v + l * 64, NHi);
    hin = X;
    K = 64;
  }

  pool_acc<<<cdiv(NHi, 256), 256, 0, stream>>>(X, batch, gsum, gmax, gcnt, NHi);
  pool_final<<<cdiv(G * 64, 256), 256, 0, stream>>>(gsum, gmax, gcnt, pooled, G);

  // classifier hidden layer via WMMA: [G,128] @ [128,64]
  gemm_wmma_f32<<<cdiv(G, 128), 256, 0, stream>>>(pooled, 128, Wc1, tmp1, G);
  bias_relu<<<cdiv(G * 64, 256), 256, 0, stream>>>(tmp1, bc1, G * 64);
  mlp_out<<<cdiv(G * C, 256), 256, 0, stream>>>(tmp1, Wc2, bc2, (float*)d_out, G, C);
}